// Pooling_83141976916902
// MI455X (gfx1250) — compile-verified
//
#include <hip/hip_runtime.h>

// Segment-sum pooling for sorted batch ids.
// softmax over a size-1 axis == 1.0 exactly, so the reference reduces to
// out[g,c] = sum_{i: batch[i]==g} x[i,c].  W and b are mathematically dead.

typedef __attribute__((ext_vector_type(2))) float v2f;  // A/B of 16x16x4 f32 WMMA
typedef __attribute__((ext_vector_type(8))) float v8f;  // C/D 16x16 f32

#define C_DIM    128
#define NUM_SEG  10000

// offsets[g] = first row index with batch[row] >= g  (batch sorted ascending)
// offsets[NUM_SEG] = n
__global__ void seg_offsets_kernel(const int* __restrict__ batch, int n,
                                   int* __restrict__ offsets) {
    int g = blockIdx.x * blockDim.x + threadIdx.x;
    if (g > NUM_SEG) return;
    if (g == NUM_SEG) { offsets[g] = n; return; }
    int lo = 0, hi = n;
    while (lo < hi) {
        int mid = (lo + hi) >> 1;
        if (batch[mid] < g) lo = mid + 1; else hi = mid;
    }
    offsets[g] = lo;
}

// One block (8 wave32) per segment.  Wave w owns columns [16w, 16w+16).
// Reduction via V_WMMA_F32_16X16X4_F32 with all-ones A:
//   D[m,n] = sum_k B[k,n] + C[m,n]  -> running column sums in the accumulator.
// B layout (mirror of documented 16x4 A layout):
//   lanes 0-15 : N = lane,    VGPR0 = row r+0 (K=0), VGPR1 = row r+1 (K=1)
//   lanes 16-31: N = lane-16, VGPR0 = row r+2 (K=2), VGPR1 = row r+3 (K=3)
// Each 16-lane half-wave load is 64B contiguous -> coalesced.
// x is streamed exactly once -> non-temporal load hint (TH=NT).
__global__ void seg_pool_wmma_kernel(const float* __restrict__ x,
                                     const int* __restrict__ offsets,
                                     float* __restrict__ out) {
    const int g     = blockIdx.x;
    const int start = offsets[g];
    const int end   = offsets[g + 1];

    const int lane    = threadIdx.x & 31;
    const int wave    = threadIdx.x >> 5;      // 0..7
    const int colbase = wave * 16;
    const int n       = lane & 15;             // column within 16-col tile
    const int half    = lane >> 4;             // 0 or 1

    v2f a;                 // all-ones A matrix (16x4)
    a.x = 1.0f;
    a.y = 1.0f;
    v8f c0 = {};           // two independent accumulator chains
    v8f c1 = {};

    // This lane's element pointer for row (start + 2*half), column colbase+n.
    const float* p = x + (size_t)(start + 2 * half) * C_DIM + colbase + n;

    // ---- Main loop: 8 rows / iteration, fully branchless, loads issued
    // ahead of the WMMAs; two accumulators halve the D->C chain depth. ----
    const int rows  = end - start;
    const int nfull = rows >> 3;               // groups of 8 rows
    for (int i = 0; i < nfull; ++i) {
        v2f b0, b1;
        b0.x = __builtin_nontemporal_load(p);
        b0.y = __builtin_nontemporal_load(p + C_DIM);
        b1.x = __builtin_nontemporal_load(p + 4 * C_DIM);
        b1.y = __builtin_nontemporal_load(p + 5 * C_DIM);
        p += 8 * C_DIM;
        // (neg_a, A, neg_b, B, c_mod, C, reuse_a, reuse_b)
        c0 = __builtin_amdgcn_wmma_f32_16x16x4_f32(false, a, false, b0,
                                                   (short)0, c0, false, false);
        c1 = __builtin_amdgcn_wmma_f32_16x16x4_f32(false, a, false, b1,
                                                   (short)0, c1, false, false);
    }

    // ---- Tail: < 8 rows.  Branchless: clamp row to end-1 (valid whenever
    // the tail runs) and zero the contribution with a 0/1 float mask.  Loop
    // bound is wave-uniform -> EXEC all-1s at the WMMA. ----
    for (int r = start + (nfull << 3); r < end; r += 4) {
        const int ra = r + 2 * half;           // K = 0 or 2
        const int rb = ra + 1;                 // K = 1 or 3
        const int ca = (ra < end) ? ra : (end - 1);
        const int cb = (rb < end) ? rb : (end - 1);
        v2f b;
        b.x = __builtin_nontemporal_load(x + (size_t)ca * C_DIM + colbase + n)
              * (float)(ra < end);
        b.y = __builtin_nontemporal_load(x + (size_t)cb * C_DIM + colbase + n)
              * (float)(rb < end);
        c0 = __builtin_amdgcn_wmma_f32_16x16x4_f32(false, a, false, b,
                                                   (short)0, c0, false, false);
    }

    // Every row of D holds identical column sums (A all ones); row M=0 is
    // element 0 of lanes 0-15 (N = lane).  Empty segments write zeros, so
    // all of d_out is overwritten every call.
    if (lane < 16) {
        out[(size_t)g * C_DIM + colbase + n] = c0[0] + c1[0];
    }
}

extern "C" void kernel_launch(void* const* d_in, const int* in_sizes, int n_in,
                              void* d_out, int out_size, void* d_ws, size_t ws_size,
                              hipStream_t stream) {
    const float* x     = (const float*)d_in[0];
    const int*   batch = (const int*)d_in[1];   // sorted segment ids
    // d_in[2] (W) and d_in[3] (b) are mathematically unused (softmax == 1).

    const int n = in_sizes[1];                  // number of rows/nodes

    int* offsets = (int*)d_ws;                  // (NUM_SEG + 1) ints

    {
        const int threads = 256;
        const int blocks  = (NUM_SEG + 1 + threads - 1) / threads;
        seg_offsets_kernel<<<blocks, threads, 0, stream>>>(batch, n, offsets);
    }
    {
        seg_pool_wmma_kernel<<<NUM_SEG, 256, 0, stream>>>(x, offsets, (float*)d_out);
    }
}